// MGQEEmbedding_45930380264185
// MI455X (gfx1250) — compile-verified
//
#include <hip/hip_runtime.h>
#include <float.h>

#define N_TOK   65536
#define SUBLEN  16
#define KHEAD   256
#define KTAIL   64
#define EMB     128
#define BN_EPS  1e-3f

typedef float v2f __attribute__((ext_vector_type(2)));
typedef float v8f __attribute__((ext_vector_type(8)));

// workspace layout (float offsets)
#define WS_HS    0      // head sum        [256]
#define WS_HSS   256    // head sum-sq     [256]
#define WS_TS    512    // tail sum        [64]
#define WS_TSS   576    // tail sum-sq     [64]
#define WS_CNT   640    // [640]=head count, [641]=tail count
#define WS_ACCSZ 642    // zeroed every call
#define WS_HA    656    // head scale      [256]
#define WS_HB    912    // head shift      [256]
#define WS_TA    1168   // tail scale      [64]
#define WS_TB    1232   // tail shift      [64]
#define WS_N2    1296   // ||c_{d,k}||^2   [8*256]
#define WS_TOTAL (WS_N2 + 8 * KHEAD)

__global__ void mgqe_zero_ws(float* __restrict__ ws) {
    int t = blockIdx.x * blockDim.x + threadIdx.x;
    if (t < WS_ACCSZ) ws[t] = 0.f;
}

// ---------------------------------------------------------------------------
// Precompute centroid squared norms once (8*256 rows of 16 floats)
// ---------------------------------------------------------------------------
__global__ __launch_bounds__(256)
void mgqe_cent_norms(const float* __restrict__ cent, float* __restrict__ ws) {
    int t = blockIdx.x * 256 + threadIdx.x;          // t = d*256 + k
    if (t >= 8 * KHEAD) return;
    const float* row = cent + (size_t)t * SUBLEN;
    float s = 0.f;
    #pragma unroll
    for (int q = 0; q < 4; ++q) {
        float4 v = *(const float4*)(row + q * 4);
        s += v.x * v.x + v.y * v.y + v.z * v.z + v.w * v.w;
    }
    ws[WS_N2 + t] = s;
}

// ---------------------------------------------------------------------------
// Pass 1: accumulate per-k BN statistics (head over all 256 k, tail over k<64)
// one wave handles (16-token tile, subspace d); 8 waves/block = all d of a tile
// ---------------------------------------------------------------------------
__global__ __launch_bounds__(256)
void mgqe_stats(const int* __restrict__ ids, const int* __restrict__ parts,
                const float* __restrict__ emb, const float* __restrict__ cent,
                float* __restrict__ ws)
{
    __shared__ float acc[WS_ACCSZ];
    for (int t = threadIdx.x; t < WS_ACCSZ; t += 256) acc[t] = 0.f;
    __syncthreads();

    const int  wave = threadIdx.x >> 5;
    const int  lane = threadIdx.x & 31;
    const int  tl   = lane & 15;
    const bool hi   = lane >= 16;
    const int  d    = wave;
    const int  t0   = blockIdx.x * 16;
    const int  tok  = t0 + tl;
    const int  koff = hi ? 2 : 0;

    const int id = ids[tok];
    const float* xrow = emb + (size_t)id * EMB + d * SUBLEN;
    const float* n2t  = ws + WS_N2 + d * KHEAD;

    // A fragments for the 4 chained K=4 WMMA steps (fp32 16x16x4 layout)
    v2f a0 = *(const v2f*)(xrow + 0  + koff);
    v2f a1 = *(const v2f*)(xrow + 4  + koff);
    v2f a2 = *(const v2f*)(xrow + 8  + koff);
    v2f a3 = *(const v2f*)(xrow + 12 + koff);

    float pa = a0.x*a0.x + a0.y*a0.y + a1.x*a1.x + a1.y*a1.y
             + a2.x*a2.x + a2.y*a2.y + a3.x*a3.x + a3.y*a3.y;
    float n1 = pa + __shfl_xor(pa, 16, 32);                 // ||x_{n,d}||^2
    float wh = (parts[tok] == 0) ? 1.f : 0.f;               // head mask

    // broadcast to match C-tile layout: VGPR i <-> token rows i / i+8
    float n1b[8], wb[8];
    #pragma unroll
    for (int i = 0; i < 8; ++i) {
        int src = hi ? (i + 8) : i;
        n1b[i] = __shfl(n1, src, 32);
        wb[i]  = __shfl(wh, src, 32);
    }

    if (wave == 0 && !hi) {  // count tokens once
        atomicAdd(&acc[WS_CNT + 0], wh);
        atomicAdd(&acc[WS_CNT + 1], 1.f - wh);
    }

    for (int cb = 0; cb < 16; ++cb) {
        const int k = cb * 16 + tl;
        const float* crow = cent + ((size_t)d * KHEAD + k) * SUBLEN;
        v2f b0 = *(const v2f*)(crow + 0  + koff);
        v2f b1 = *(const v2f*)(crow + 4  + koff);
        v2f b2 = *(const v2f*)(crow + 8  + koff);
        v2f b3 = *(const v2f*)(crow + 12 + koff);
        float n2 = n2t[k];                                   // precomputed
        if (cb < 15) __builtin_prefetch(crow + 16 * SUBLEN, 0, 3);

        v8f c = {};
        c = __builtin_amdgcn_wmma_f32_16x16x4_f32(false, a0, false, b0, (short)0, c, false, false);
        c = __builtin_amdgcn_wmma_f32_16x16x4_f32(false, a1, false, b1, (short)0, c, false, false);
        c = __builtin_amdgcn_wmma_f32_16x16x4_f32(false, a2, false, b2, (short)0, c, false, false);
        c = __builtin_amdgcn_wmma_f32_16x16x4_f32(false, a3, false, b3, (short)0, c, false, false);

        float shs = 0.f, shss = 0.f, sts = 0.f, stss = 0.f;
        #pragma unroll
        for (int i = 0; i < 8; ++i) {
            float r = 2.f * c[i] - n1b[i] - n2;
            float w = wb[i];
            shs  += w * r;
            shss += w * r * r;
            float t = 1.f - w;
            sts  += t * r;
            stss += t * r * r;
        }
        shs  += __shfl_xor(shs, 16, 32);
        shss += __shfl_xor(shss, 16, 32);
        if (!hi) {
            atomicAdd(&acc[WS_HS  + k], shs);
            atomicAdd(&acc[WS_HSS + k], shss);
        }
        if (cb < 4) {
            sts  += __shfl_xor(sts, 16, 32);
            stss += __shfl_xor(stss, 16, 32);
            if (!hi) {
                atomicAdd(&acc[WS_TS  + k], sts);
                atomicAdd(&acc[WS_TSS + k], stss);
            }
        }
    }
    __syncthreads();
    for (int t = threadIdx.x; t < WS_ACCSZ; t += 256)
        atomicAdd(&ws[t], acc[t]);
}

// ---------------------------------------------------------------------------
// Turn sums into per-k affine (scale, shift): norm(r) = a_k * r + b_k
// ---------------------------------------------------------------------------
__global__ void mgqe_finalize(float* __restrict__ ws) {
    int k = threadIdx.x;
    float cntH = fmaxf(ws[WS_CNT + 0] * 8.f, 1.f);
    float cntT = fmaxf(ws[WS_CNT + 1] * 8.f, 1.f);
    if (k < KHEAD) {
        float m = ws[WS_HS + k] / cntH;
        float v = ws[WS_HSS + k] / cntH - m * m;
        float a = rsqrtf(v + BN_EPS);
        ws[WS_HA + k] = a;
        ws[WS_HB + k] = -m * a;
    }
    if (k < KTAIL) {
        float m = ws[WS_TS + k] / cntT;
        float v = ws[WS_TSS + k] / cntT - m * m;
        float a = rsqrtf(v + BN_EPS);
        ws[WS_TA + k] = a;
        ws[WS_TB + k] = -m * a;
    }
}

// ---------------------------------------------------------------------------
// Pass 2: recompute responses via WMMA, normalize, argmax, gather centroid out
// ---------------------------------------------------------------------------
__global__ __launch_bounds__(256)
void mgqe_codes_out(const int* __restrict__ ids, const int* __restrict__ parts,
                    const float* __restrict__ emb, const float* __restrict__ cent,
                    const float* __restrict__ ws, float* __restrict__ out)
{
    __shared__ int codes[8][16];

    const int  wave = threadIdx.x >> 5;
    const int  lane = threadIdx.x & 31;
    const int  tl   = lane & 15;
    const bool hi   = lane >= 16;
    const int  d    = wave;
    const int  t0   = blockIdx.x * 16;
    const int  tok  = t0 + tl;
    const int  koff = hi ? 2 : 0;

    const int id = ids[tok];
    const float* xrow = emb + (size_t)id * EMB + d * SUBLEN;
    const float* n2t  = ws + WS_N2 + d * KHEAD;

    v2f a0 = *(const v2f*)(xrow + 0  + koff);
    v2f a1 = *(const v2f*)(xrow + 4  + koff);
    v2f a2 = *(const v2f*)(xrow + 8  + koff);
    v2f a3 = *(const v2f*)(xrow + 12 + koff);

    float pa = a0.x*a0.x + a0.y*a0.y + a1.x*a1.x + a1.y*a1.y
             + a2.x*a2.x + a2.y*a2.y + a3.x*a3.x + a3.y*a3.y;
    float n1 = pa + __shfl_xor(pa, 16, 32);
    float wh = (parts[tok] == 0) ? 1.f : 0.f;

    float n1b[8], wb[8];
    #pragma unroll
    for (int i = 0; i < 8; ++i) {
        int src = hi ? (i + 8) : i;
        n1b[i] = __shfl(n1, src, 32);
        wb[i]  = __shfl(wh, src, 32);
    }

    float best[8];
    int   bestk[8];
    #pragma unroll
    for (int i = 0; i < 8; ++i) { best[i] = -FLT_MAX; bestk[i] = 0; }

    for (int cb = 0; cb < 16; ++cb) {
        const int k = cb * 16 + tl;
        const float* crow = cent + ((size_t)d * KHEAD + k) * SUBLEN;
        v2f b0 = *(const v2f*)(crow + 0  + koff);
        v2f b1 = *(const v2f*)(crow + 4  + koff);
        v2f b2 = *(const v2f*)(crow + 8  + koff);
        v2f b3 = *(const v2f*)(crow + 12 + koff);
        float n2 = n2t[k];                                   // precomputed
        if (cb < 15) __builtin_prefetch(crow + 16 * SUBLEN, 0, 3);

        float ha  = ws[WS_HA + k];
        float hbv = ws[WS_HB + k];
        bool  tv  = (cb < 4);
        float ta  = tv ? ws[WS_TA + k] : 0.f;
        float tbv = tv ? ws[WS_TB + k] : 0.f;

        v8f c = {};
        c = __builtin_amdgcn_wmma_f32_16x16x4_f32(false, a0, false, b0, (short)0, c, false, false);
        c = __builtin_amdgcn_wmma_f32_16x16x4_f32(false, a1, false, b1, (short)0, c, false, false);
        c = __builtin_amdgcn_wmma_f32_16x16x4_f32(false, a2, false, b2, (short)0, c, false, false);
        c = __builtin_amdgcn_wmma_f32_16x16x4_f32(false, a3, false, b3, (short)0, c, false, false);

        #pragma unroll
        for (int i = 0; i < 8; ++i) {
            float r  = 2.f * c[i] - n1b[i] - n2;
            float sc = (wb[i] != 0.f) ? (ha * r + hbv)
                                      : (tv ? (ta * r + tbv) : -FLT_MAX);
            if (sc > best[i]) { best[i] = sc; bestk[i] = k; }
        }
    }

    // butterfly argmax across the 16 centroid lanes of each half-wave
    #pragma unroll
    for (int m = 8; m >= 1; m >>= 1) {
        #pragma unroll
        for (int i = 0; i < 8; ++i) {
            float ov = __shfl_xor(best[i], m, 32);
            int   ok = __shfl_xor(bestk[i], m, 32);
            if (ov > best[i] || (ov == best[i] && ok < bestk[i])) {
                best[i] = ov; bestk[i] = ok;
            }
        }
    }

    if (lane == 0) {
        #pragma unroll
        for (int i = 0; i < 8; ++i) codes[wave][i] = bestk[i];
    }
    if (lane == 16) {
        #pragma unroll
        for (int i = 0; i < 8; ++i) codes[wave][i + 8] = bestk[i];
    }
    __syncthreads();

    // gather the winning centroid row and stream it out (float4 stores)
    const int code = codes[wave][tl];
    const float* orow = cent + ((size_t)d * KHEAD + code) * SUBLEN;
    const int off = hi ? 8 : 0;
    float4 va = *(const float4*)(orow + off);
    float4 vb = *(const float4*)(orow + off + 4);
    float* dst = out + (size_t)tok * EMB + d * SUBLEN + off;
    *(float4*)(dst)     = va;
    *(float4*)(dst + 4) = vb;
}

extern "C" void kernel_launch(void* const* d_in, const int* in_sizes, int n_in,
                              void* d_out, int out_size, void* d_ws, size_t ws_size,
                              hipStream_t stream) {
    (void)in_sizes; (void)n_in; (void)out_size; (void)ws_size;
    const int*   ids   = (const int*)d_in[0];    // inputs  (1024,64) int32
    const int*   parts = (const int*)d_in[1];    // partitions (65536,) int32
    const float* emb   = (const float*)d_in[2];  // emb_table (100000,128) f32
    const float* cent  = (const float*)d_in[3];  // centroids (8,256,16) f32
    float* out = (float*)d_out;
    float* ws  = (float*)d_ws;

    mgqe_zero_ws<<<(WS_ACCSZ + 255) / 256, 256, 0, stream>>>(ws);
    mgqe_cent_norms<<<(8 * KHEAD) / 256, 256, 0, stream>>>(cent, ws);
    mgqe_stats<<<N_TOK / 16, 256, 0, stream>>>(ids, parts, emb, cent, ws);
    mgqe_finalize<<<1, 256, 0, stream>>>(ws);
    mgqe_codes_out<<<N_TOK / 16, 256, 0, stream>>>(ids, parts, emb, cent, ws, out);
}